// JPEG_Coef2RGB_1924145349323
// MI455X (gfx1250) — compile-verified
//
#include <hip/hip_runtime.h>

typedef __attribute__((ext_vector_type(2))) float v2f;
typedef __attribute__((ext_vector_type(8))) float v8f;

#define CHROMA_PITCH 68   // 32x64 chroma plane, padded pitch (floats)

// IDCT basis: M[k=8u+v][c=8x+y] = 0.25*a(u)a(v)cos((2x+1)u pi/16)cos((2y+1)v pi/16)
__device__ __forceinline__ float idct_m(int k, int c) {
  const float PI16 = 0.19634954084936207f;  // pi/16
  int u = k >> 3, v = k & 7;
  int x = c >> 3, y = c & 7;
  float au = (u == 0) ? 0.70710678118654752f : 1.0f;
  float av = (v == 0) ? 0.70710678118654752f : 1.0f;
  return 0.25f * au * av * cosf((float)((2 * x + 1) * u) * PI16)
                         * cosf((float)((2 * y + 1) * v) * PI16);
}

__device__ __forceinline__ float clampf(float a, float lo, float hi) {
  return fminf(fmaxf(a, lo), hi);
}

// One workgroup = 64 (rows) x 128 (cols) output pixels of one image.
//   waves 0..7 : Y GEMM, 16 blocks each (block-row w, 16 block-cols)
//   waves 8..9 : U GEMM, 16 blocks each (2 chroma block-rows x 8 cols)
//   waves 10..11: V GEMM likewise
// Then U/V land in LDS planes; Y waves fuse depth_to_space + clip + upsample
// + YUV->RGB straight out of their WMMA accumulators.
__global__ __launch_bounds__(384)
void jpeg_coef2rgb_kernel(const float* __restrict__ ydcc,
                          const float* __restrict__ udcc,
                          const float* __restrict__ vdcc,
                          float* __restrict__ out) {
  __shared__ v2f   sM[16 * 4 * 32];          // IDCT matrix, per-lane B-fragment layout
  __shared__ float sU[32 * CHROMA_PITCH];    // decoded chroma planes for this tile
  __shared__ float sV[32 * CHROMA_PITCH];

  const int tid  = threadIdx.x;
  const int wv   = tid >> 5;        // wave 0..11 (wave32)
  const int lane = tid & 31;
  const int h    = lane >> 4;       // lane half: selects K pair (A) / K rows (B)
  const int Lm   = lane & 15;       // matrix row (A) / column (B,C,D)
  const int tc   = blockIdx.x;      // 0..7   (128-px column tile)
  const int tr   = blockIdx.y;      // 0..15  (64-px row tile)
  const int bz   = blockIdx.z;      // 0..7   batch

  // ---- Build swizzled IDCT matrix in LDS -------------------------------
  // sM[(kk*4+nt)*32 + lane] = { M[4kk+2h][16nt+Lm], M[4kk+2h+1][16nt+Lm] }
  for (int e = tid; e < 2048; e += 384) {
    int l  = e & 31, q = e >> 5;
    int nt = q & 3,  kk = q >> 2;
    int hh = l >> 4, ll = l & 15;
    int c  = nt * 16 + ll;
    int k0 = kk * 4 + hh * 2;
    v2f m;
    m.x = idct_m(k0, c);
    m.y = idct_m(k0 + 1, c);
    sM[e] = m;
  }
  __syncthreads();

  // ---- Per-wave A base (16 blocks = 16 GEMM rows) ----------------------
  const float* abase;
  if (wv < 8) {
    int by = tr * 8 + wv;           // Y block row (0..127)
    int bx = tc * 16 + Lm;          // Y block col (0..127)
    abase = ydcc + ((((size_t)bz * 128 + by) * 128 + bx) << 6);
  } else {
    const float* src = (wv < 10) ? udcc : vdcc;
    int crb = ((wv - 8) & 1) * 2;   // chroma block-row base within tile: 0 or 2
    int ch  = tr * 4 + crb + (Lm >> 3);
    int cw  = tc * 8 + (Lm & 7);
    abase = src + ((((size_t)bz * 64 + ch) * 64 + cw) << 6);
  }
  abase += h * 2;                   // lanes 16..31 supply K=2,3 of each 4-step

  // A fragments: 16 x float2 per lane (whole 16x64 slab in registers)
  v2f af[16];
#pragma unroll
  for (int kk = 0; kk < 16; ++kk)
    af[kk] = *(const v2f*)(abase + kk * 4);

  v8f acc[4];
#pragma unroll
  for (int nt = 0; nt < 4; ++nt)
    acc[nt] = (v8f){0.f, 0.f, 0.f, 0.f, 0.f, 0.f, 0.f, 0.f};

  // ---- GEMM: C[16x64] += A[16x64] * M[64x64], K-step = 4 ---------------
#pragma unroll
  for (int kk = 0; kk < 16; ++kk) {
#pragma unroll
    for (int nt = 0; nt < 4; ++nt) {
      v2f bf = sM[(kk * 4 + nt) * 32 + lane];
      acc[nt] = __builtin_amdgcn_wmma_f32_16x16x4_f32(
          false, af[kk], false, bf, (short)0, acc[nt], false, false);
    }
  }

  // ---- U/V waves: scatter decoded chroma into LDS planes ---------------
  if (wv >= 8) {
    float* plane = (wv < 10) ? sU : sV;
    int crb = ((wv - 8) & 1) * 2;
#pragma unroll
    for (int nt = 0; nt < 4; ++nt) {
#pragma unroll
      for (int r = 0; r < 8; ++r) {
        // C element (row m=r+8h, col c=16nt+Lm); c=8x+y -> pixel (x,y) in block
        int x = nt * 2 + (Lm >> 3);
        int y = Lm & 7;
        int row = (crb + h) * 8 + x;   // chroma-plane row (0..31)
        int col = r * 8 + y;           // chroma-plane col (0..63): cc = m&7 = r
        plane[row * CHROMA_PITCH + col] = acc[nt][r];
      }
    }
  }
  __syncthreads();

  // ---- Y waves: fused depth_to_space + clip + upsample + YUV->RGB ------
  if (wv < 8) {
#pragma unroll
    for (int nt = 0; nt < 4; ++nt) {
      int crow = wv * 4 + nt;          // (w*8 + x)>>1 is nt-uniform
#pragma unroll
      for (int r = 0; r < 8; ++r) {
        int m = r + 8 * h;             // block index within wave's 16-block slab
        int x = nt * 2 + (Lm >> 3);
        int y = Lm & 7;
        float yv = clampf(acc[nt][r] + 0.5f, 0.0f, 1.0f);
        int ccol = m * 4 + (y >> 1);
        float uu = clampf(sU[crow * CHROMA_PITCH + ccol], -0.5f, 0.5f);
        float vvv = clampf(sV[crow * CHROMA_PITCH + ccol], -0.5f, 0.5f);
        float R = clampf(yv + 1.13988303f * vvv, 0.0f, 1.0f);
        float G = clampf(yv - 0.394642334f * uu - 0.58062185f * vvv, 0.0f, 1.0f);
        float B = clampf(yv + 2.03206185f * uu, 0.0f, 1.0f);
        int grow = tr * 64 + wv * 8 + x;
        int gcol = tc * 128 + m * 8 + y;
        size_t o = ((((size_t)bz * 1024 + grow) * 1024) + gcol) * 3;
        out[o + 0] = R;
        out[o + 1] = G;
        out[o + 2] = B;
      }
    }
  }
}

extern "C" void kernel_launch(void* const* d_in, const int* in_sizes, int n_in,
                              void* d_out, int out_size, void* d_ws, size_t ws_size,
                              hipStream_t stream) {
  (void)in_sizes; (void)n_in; (void)d_ws; (void)ws_size; (void)out_size;
  const float* ydcc = (const float*)d_in[0];  // [8,128,128,64]
  const float* udcc = (const float*)d_in[1];  // [8,64,64,64]
  const float* vdcc = (const float*)d_in[2];  // [8,64,64,64]
  float* out = (float*)d_out;                 // [8,1024,1024,3]
  dim3 grid(8, 16, 8);                        // col tiles, row tiles, batch
  jpeg_coef2rgb_kernel<<<grid, dim3(384, 1, 1), 0, stream>>>(ydcc, udcc, vdcc, out);
}